// GTLayer_42614665511412
// MI455X (gfx1250) — compile-verified
//
#include <hip/hip_runtime.h>

#define NN   4096
#define KCH  5
#define OCH  2
#define NE   131072
#define LDSS 18   // LDS row pad: 18 floats = 72B, 8B-aligned rows, conflict-free b64

typedef float v2f __attribute__((ext_vector_type(2)));
typedef float v8f __attribute__((ext_vector_type(8)));

// ---------------------------------------------------------------- zero scratch
__global__ void zero_kernel(float4* __restrict__ p, long nvec) {
    long i      = (long)blockIdx.x * blockDim.x + threadIdx.x;
    long stride = (long)gridDim.x * blockDim.x;
    const float4 z = make_float4(0.f, 0.f, 0.f, 0.f);
    for (; i < nvec; i += stride) p[i] = z;
}

// ---------------------------------------------------------------- softmax(w[o,:])[k]
__device__ __forceinline__ float softmax_w(const float* __restrict__ w, int o, int k) {
    float mx = w[o * KCH];
#pragma unroll
    for (int j = 1; j < KCH; ++j) mx = fmaxf(mx, w[o * KCH + j]);
    float sum = 0.f;
#pragma unroll
    for (int j = 0; j < KCH; ++j) sum += __expf(w[o * KCH + j] - mx);
    return __expf(w[o * KCH + k] - mx) / sum;
}

// ---------------------------------------------------------------- W1/W2 outputs
__global__ void weights_kernel(const float* __restrict__ w1, const float* __restrict__ w2,
                               float* __restrict__ o1, float* __restrict__ o2) {
    int t = threadIdx.x;
    if (t < OCH * KCH) {
        int o = t / KCH, k = t % KCH;
        o1[t] = softmax_w(w1, o, k);
        o2[t] = softmax_w(w2, o, k);
    }
}

// ---------------------------------------------------------------- edge scatter-add
__global__ void scatter_kernel(const int* __restrict__ ei, const float* __restrict__ ev,
                               const float* __restrict__ w1, const float* __restrict__ w2,
                               float* __restrict__ A1, float* __restrict__ A2) {
    int idx = blockIdx.x * blockDim.x + threadIdx.x;
    const int total = OCH * KCH * NE;
    if (idx >= total) return;
    int e = idx % NE;
    int k = (idx / NE) % KCH;
    int o = idx / (NE * KCH);

    int   row = ei[(k * 2 + 0) * NE + e];
    int   col = ei[(k * 2 + 1) * NE + e];
    float v   = ev[k * NE + e];

    size_t off = ((size_t)o * NN + row) * NN + col;
    atomicAdd(&A1[off], softmax_w(w1, o, k) * v);
    atomicAdd(&A2[off], softmax_w(w2, o, k) * v);
}

// ---------------------------------------------------------------- batched GEMM (fp32 WMMA)
// Block tile 128x128, BK=16, 256 threads (8 waves, 4x2), wave tile 32x64 (2x4 WMMA tiles).
__global__ __launch_bounds__(256) void gemm_kernel(const float* __restrict__ Ag,
                                                   const float* __restrict__ Bg,
                                                   float* __restrict__ Cg) {
    __shared__ float As[128][LDSS];   // As[m][k], row-major
    __shared__ float Bs[128][LDSS];   // Bs[n][k], B stored transposed

    const int tid   = threadIdx.x;
    const int lane  = tid & 31;
    const int wave  = tid >> 5;
    const int laneR = lane & 15;      // N / M index within a 16-wide fragment
    const int laneH = lane >> 4;      // half-wave select: K pair 0/1 vs 2/3
    const int waveM = (wave >> 1) * 32;
    const int waveN = (wave & 1) * 64;
    const int m0    = blockIdx.y * 128;
    const int n0    = blockIdx.x * 128;

    const size_t chOff = (size_t)blockIdx.z * NN * NN;
    const float* __restrict__ A = Ag + chOff;
    const float* __restrict__ B = Bg + chOff;
    float* __restrict__ C       = Cg + chOff;

    v8f acc[2][4] = {};

    for (int k0 = 0; k0 < NN; k0 += 16) {
        // ---- stage tiles into LDS: 2 x float4 of A and B per thread
#pragma unroll
        for (int i = 0; i < 2; ++i) {
            const int v  = tid + i * 256;          // 0..511
            const int ar = v >> 2;                 // 0..127 (A row)
            const int ac = (v & 3) << 2;           // 0,4,8,12 (A col vec)
            const float4 a4 = *(const float4*)(A + (size_t)(m0 + ar) * NN + k0 + ac);
            *(float2*)&As[ar][ac]     = make_float2(a4.x, a4.y);
            *(float2*)&As[ar][ac + 2] = make_float2(a4.z, a4.w);

            const int bk = v >> 5;                 // 0..15  (B row = k)
            const int bn = (v & 31) << 2;          // 0..124 (B col vec = n)
            const float4 b4 = *(const float4*)(B + (size_t)(k0 + bk) * NN + n0 + bn);
            Bs[bn + 0][bk] = b4.x;
            Bs[bn + 1][bk] = b4.y;
            Bs[bn + 2][bk] = b4.z;
            Bs[bn + 3][bk] = b4.w;
        }
        // prefetch next k-chunk (gfx1250 global_prefetch_b8 path)
        if (k0 + 16 < NN) {
            __builtin_prefetch(A + (size_t)(m0 + (tid >> 1)) * NN + k0 + 16, 0, 1);
            __builtin_prefetch(B + (size_t)(k0 + 16 + (tid >> 4)) * NN + n0 + ((tid & 15) << 3), 0, 1);
        }
        __syncthreads();

        // ---- 4 WMMA k-steps of K=4 each over the 16-wide chunk
#pragma unroll
        for (int kk = 0; kk < 16; kk += 4) {
            const int koff = kk + (laneH << 1);    // lanes 0-15: kk,kk+1 ; lanes 16-31: kk+2,kk+3
            v2f aF[2];
#pragma unroll
            for (int i = 0; i < 2; ++i)
                aF[i] = *(const v2f*)&As[waveM + i * 16 + laneR][koff];
            v2f bF[4];
#pragma unroll
            for (int j = 0; j < 4; ++j)
                bF[j] = *(const v2f*)&Bs[waveN + j * 16 + laneR][koff];
#pragma unroll
            for (int i = 0; i < 2; ++i)
#pragma unroll
                for (int j = 0; j < 4; ++j)
                    acc[i][j] = __builtin_amdgcn_wmma_f32_16x16x4_f32(
                        false, aF[i], false, bF[j], (short)0, acc[i][j], false, false);
        }
        __syncthreads();
    }

    // ---- write back: C/D layout — VGPR v holds M=v (lanes 0-15) / M=v+8 (lanes 16-31)
#pragma unroll
    for (int i = 0; i < 2; ++i) {
#pragma unroll
        for (int j = 0; j < 4; ++j) {
            const int colg = n0 + waveN + j * 16 + laneR;
#pragma unroll
            for (int v = 0; v < 8; ++v) {
                const int rowg = m0 + waveM + i * 16 + v + (laneH << 3);
                C[(size_t)rowg * NN + colg] = acc[i][j][v];
            }
        }
    }
}

// ----------------------------------------------------------------
extern "C" void kernel_launch(void* const* d_in, const int* in_sizes, int n_in,
                              void* d_out, int out_size, void* d_ws, size_t ws_size,
                              hipStream_t stream) {
    (void)in_sizes; (void)n_in; (void)out_size; (void)ws_size;

    const int*   edge_index = (const int*)d_in[0];    // [K,2,E]
    const float* edge_value = (const float*)d_in[1];  // [K,E]
    const float* w1         = (const float*)d_in[2];  // [O,K]
    const float* w2         = (const float*)d_in[3];  // [O,K]

    float* H   = (float*)d_out;                       // [O,N,N]
    float* oW1 = H + (size_t)OCH * NN * NN;           // [O,K]
    float* oW2 = oW1 + OCH * KCH;                     // [O,K]

    float* A1 = (float*)d_ws;                         // [O,N,N]
    float* A2 = A1 + (size_t)OCH * NN * NN;           // [O,N,N]

    // 1) zero A1/A2 (256 MB of workspace)
    const long nvec = (long)2 * OCH * NN * NN / 4;
    zero_kernel<<<4096, 256, 0, stream>>>((float4*)d_ws, nvec);

    // 2) scatter-add weighted edges into both adjacencies
    const int total = OCH * KCH * NE;
    scatter_kernel<<<(total + 255) / 256, 256, 0, stream>>>(edge_index, edge_value,
                                                            w1, w2, A1, A2);

    // 3) softmax weight outputs
    weights_kernel<<<1, 32, 0, stream>>>(w1, w2, oW1, oW2);

    // 4) H[o] = A1[o] @ A2[o] via V_WMMA_F32_16X16X4_F32
    dim3 grid(NN / 128, NN / 128, OCH);
    gemm_kernel<<<grid, 256, 0, stream>>>(A1, A2, H);
}